// MultiHeadAttentionLayer_33088428049200
// MI455X (gfx1250) — compile-verified
//
#include <hip/hip_runtime.h>
#include <hip/hip_bf16.h>

// Graph-transformer attention layer for MI455X (gfx1250, wave32).
// Fixed sizes from reference: N=50000 (mult of 16), E=1600000 (mult of 16),
// IN_DIM=64, H*D=64.
//
// Roofline: traffic dominated by streaming e (410MB) + e_out (410MB); Q/K/V
// (38MB) and the h_out/z accumulators (13MB) are kept L2-resident by marking
// the streams non-temporal. GEMMs run on V_WMMA_F32_16X16X4_F32 (exact fp32).

typedef float v2f __attribute__((ext_vector_type(2)));
typedef float v4f __attribute__((ext_vector_type(4)));
typedef float v8f __attribute__((ext_vector_type(8)));

#define LSTR 68   // LDS row stride in floats (64 + 4) to avoid bank conflicts

// ---------------------------------------------------------------------------
// Node projection: out[16-row tile] = h_tile @ W + b, via V_WMMA_F32_16X16X4_F32
// Block: 128 threads = 4 waves; wave ct computes output columns [16ct,16ct+16).
// ---------------------------------------------------------------------------
__global__ __launch_bounds__(128)
void node_proj_kernel(const float* __restrict__ h,
                      const float* __restrict__ W,
                      const float* __restrict__ b,
                      float* __restrict__ out) {
  __shared__ float tile[16 * LSTR];

  const int tid = threadIdx.x;
  const long rowbase = (long)blockIdx.x * 16;

  // Stage 16x64 input tile into LDS (padded rows). 128 threads x 8 floats.
  {
    const int r  = tid >> 3;
    const int cb = (tid & 7) * 8;
    const v4f* s4 = (const v4f*)(h + (rowbase + r) * 64 + cb);
    v4f x0 = s4[0];
    v4f x1 = s4[1];
    *(v4f*)(&tile[r * LSTR + cb])     = x0;
    *(v4f*)(&tile[r * LSTR + cb + 4]) = x1;
  }
  __syncthreads();

  const int lane = tid & 31;
  const int ct   = tid >> 5;              // wave id -> col tile
  const int col  = ct * 16 + (lane & 15);
  const int koff = (lane >> 4) * 2;       // lanes 16-31 hold K+2,K+3
  const int arow = lane & 15;

  const float bias = b[col];
  v8f acc;
#pragma unroll
  for (int i = 0; i < 8; ++i) acc[i] = bias;

#pragma unroll
  for (int kk = 0; kk < 16; ++kk) {
    const int k0 = kk * 4 + koff;
    // A fragment 16x4 f32: lane&15 = M row, VGPRs stripe K
    v2f a = *(const v2f*)(&tile[arow * LSTR + k0]);
    // B fragment 4x16 f32: lane&15 = N col, VGPRs stripe K
    v2f bb;
    bb.x = W[(k0    ) * 64 + col];
    bb.y = W[(k0 + 1) * 64 + col];
    acc = __builtin_amdgcn_wmma_f32_16x16x4_f32(
        false, a, false, bb, (short)0, acc, false, false);
  }

  // C/D layout: VGPR i -> row i (lanes 0-15) / row i+8 (lanes 16-31)
  const int orow = (lane >> 4) * 8;
#pragma unroll
  for (int i = 0; i < 8; ++i) {
    out[(rowbase + orow + i) * 64 + col] = acc[i];
  }
}

// ---------------------------------------------------------------------------
// Edge kernel: pe = e_tile @ We + be (WMMA), then fused elementwise score,
// e_out store (non-temporal), s = exp(clip(sum)), atomic scatter into the
// h-accumulator and z (L2-resident). Block: 128 threads, 16 edges per block.
// ---------------------------------------------------------------------------
__global__ __launch_bounds__(128)
void edge_kernel(const float* __restrict__ e,
                 const float* __restrict__ We,
                 const float* __restrict__ be,
                 const float* __restrict__ Qn,
                 const float* __restrict__ Kn,
                 const float* __restrict__ Vn,
                 const int* __restrict__ src,
                 const int* __restrict__ dst,
                 float* __restrict__ hacc,
                 float* __restrict__ z,
                 float* __restrict__ e_out) {
  __shared__ float etile[16 * LSTR];
  __shared__ float pe[16 * LSTR];

  const int tid = threadIdx.x;
  const long ebase = (long)blockIdx.x * 16;

  // Prefetch next tile of e while we work on this one (speculative).
  if (blockIdx.x + 1 < gridDim.x && tid < 16) {
    __builtin_prefetch(e + (ebase + 16 + tid) * 64, 0, 1);
  }

  // Stage 16x64 edge-feature tile into LDS. e is streamed once -> NT loads
  // so it bypasses long-term L2 residency (keep Q/K/V + accumulators hot).
  {
    const int r  = tid >> 3;
    const int cb = (tid & 7) * 8;
    const v4f* s4 = (const v4f*)(e + (ebase + r) * 64 + cb);
    v4f x0 = __builtin_nontemporal_load(s4);
    v4f x1 = __builtin_nontemporal_load(s4 + 1);
    *(v4f*)(&etile[r * LSTR + cb])     = x0;
    *(v4f*)(&etile[r * LSTR + cb + 4]) = x1;
  }
  __syncthreads();

  // ---- pe = e_tile @ We + be via WMMA f32 16x16x4 ----
  {
    const int lane = tid & 31;
    const int ct   = tid >> 5;
    const int col  = ct * 16 + (lane & 15);
    const int koff = (lane >> 4) * 2;
    const int arow = lane & 15;

    const float bias = be[col];
    v8f acc;
#pragma unroll
    for (int i = 0; i < 8; ++i) acc[i] = bias;

#pragma unroll
    for (int kk = 0; kk < 16; ++kk) {
      const int k0 = kk * 4 + koff;
      v2f a = *(const v2f*)(&etile[arow * LSTR + k0]);
      v2f bb;
      bb.x = We[(k0    ) * 64 + col];
      bb.y = We[(k0 + 1) * 64 + col];
      acc = __builtin_amdgcn_wmma_f32_16x16x4_f32(
          false, a, false, bb, (short)0, acc, false, false);
    }
    const int orow = (lane >> 4) * 8;
#pragma unroll
    for (int i = 0; i < 8; ++i) {
      pe[(orow + i) * LSTR + col] = acc[i];
    }
  }
  __syncthreads();

  // ---- per-(edge, head) elementwise phase: thread = (edge_local, head) ----
  const int el   = tid >> 3;   // 0..15
  const int head = tid & 7;    // 0..7
  const long eg  = ebase + el;

  // Index streams are read exactly once -> non-temporal.
  const int s = __builtin_nontemporal_load(src + eg);
  const int d = __builtin_nontemporal_load(dst + eg);

  // Q/K/V gathers: default (RT) policy, L2-resident working set.
  const v4f* kp = (const v4f*)(Kn + (long)s * 64 + head * 8);
  const v4f* qp = (const v4f*)(Qn + (long)d * 64 + head * 8);
  const v4f* vp = (const v4f*)(Vn + (long)s * 64 + head * 8);
  v4f k0 = kp[0], k1 = kp[1];
  v4f q0 = qp[0], q1 = qp[1];
  v4f v0 = vp[0], v1 = vp[1];
  v4f p0 = *(const v4f*)(&pe[el * LSTR + head * 8]);
  v4f p1 = *(const v4f*)(&pe[el * LSTR + head * 8 + 4]);

  const float inv = 0.35355339059327373f;  // 1/sqrt(8)

  v4f sc0 = k0 * q0 * inv * p0;
  v4f sc1 = k1 * q1 * inv * p1;

  // e_out (pre-exp score): streamed once -> non-temporal stores, fully
  // coalesced (8 threads cover one 256B row, 16 rows contiguous per block).
  v4f* op = (v4f*)(e_out + eg * 64 + head * 8);
  __builtin_nontemporal_store(sc0, op);
  __builtin_nontemporal_store(sc1, op + 1);

  float sum = ((sc0.x + sc0.y) + (sc0.z + sc0.w)) +
              ((sc1.x + sc1.y) + (sc1.z + sc1.w));
  sum = fminf(fmaxf(sum, -5.0f), 5.0f);
  const float sv = expf(sum);

  // Scatter: hacc[d, head, :] += V[s, head, :] * sv ; z[d, head] += sv
  float* hp = hacc + (long)d * 64 + head * 8;
  unsafeAtomicAdd(hp + 0, v0.x * sv);
  unsafeAtomicAdd(hp + 1, v0.y * sv);
  unsafeAtomicAdd(hp + 2, v0.z * sv);
  unsafeAtomicAdd(hp + 3, v0.w * sv);
  unsafeAtomicAdd(hp + 4, v1.x * sv);
  unsafeAtomicAdd(hp + 5, v1.y * sv);
  unsafeAtomicAdd(hp + 6, v1.z * sv);
  unsafeAtomicAdd(hp + 7, v1.w * sv);
  unsafeAtomicAdd(&z[(long)d * 8 + head], sv);
}

// ---------------------------------------------------------------------------
// Finalize: h_out = wV / (z + 1e-6)
// ---------------------------------------------------------------------------
__global__ __launch_bounds__(256)
void finalize_kernel(float* __restrict__ hout, const float* __restrict__ z,
                     int total) {
  const int i = blockIdx.x * blockDim.x + threadIdx.x;
  if (i < total) {
    const int node = i >> 6;
    const int head = (i >> 3) & 7;
    hout[i] = hout[i] / (z[node * 8 + head] + 1e-6f);
  }
}

extern "C" void kernel_launch(void* const* d_in, const int* in_sizes, int n_in,
                              void* d_out, int out_size, void* d_ws, size_t ws_size,
                              hipStream_t stream) {
  const float* h  = (const float*)d_in[0];
  const float* e  = (const float*)d_in[1];
  const float* Wq = (const float*)d_in[2];
  const float* bq = (const float*)d_in[3];
  const float* Wk = (const float*)d_in[4];
  const float* bk = (const float*)d_in[5];
  const float* Wv = (const float*)d_in[6];
  const float* bv = (const float*)d_in[7];
  const float* We = (const float*)d_in[8];
  const float* be = (const float*)d_in[9];
  const int* src  = (const int*)d_in[10];
  const int* dst  = (const int*)d_in[11];

  const int  N = in_sizes[0] / 64;        // 50000 (multiple of 16)
  const long E = (long)in_sizes[1] / 64;  // 1600000 (multiple of 16)

  float* out   = (float*)d_out;
  float* hout  = out;                   // [N,64]  (wV accumulator, then final)
  float* e_out = out + (long)N * 64;    // [E,64]

  float* ws = (float*)d_ws;
  float* Qn = ws;                       // [N,64]
  float* Kn = ws + (long)N * 64;        // [N,64]
  float* Vn = ws + 2L * (long)N * 64;   // [N,64]
  float* z  = ws + 3L * (long)N * 64;   // [N,8]

  // Zero the accumulators (graph-capture-safe async memsets).
  (void)hipMemsetAsync(hout, 0, (size_t)N * 64 * sizeof(float), stream);
  (void)hipMemsetAsync(z,    0, (size_t)N * 8  * sizeof(float), stream);

  const int ntiles = N / 16;
  node_proj_kernel<<<ntiles, 128, 0, stream>>>(h, Wq, bq, Qn);
  node_proj_kernel<<<ntiles, 128, 0, stream>>>(h, Wk, bk, Kn);
  node_proj_kernel<<<ntiles, 128, 0, stream>>>(h, Wv, bv, Vn);

  const int etiles = (int)(E / 16);
  edge_kernel<<<etiles, 128, 0, stream>>>(e, We, be, Qn, Kn, Vn, src, dst,
                                          hout, z, e_out);

  const int total = N * 64;
  finalize_kernel<<<(total + 255) / 256, 256, 0, stream>>>(hout, z, total);
}